// Latent_Generator_37460704755833
// MI455X (gfx1250) — compile-verified
//
#include <hip/hip_runtime.h>

// Mixture-of-Gaussians latent generator for MI455X (gfx1250).
// z[b] = A[k[b]] @ eps[b] + mu[k[b]],  B=4096, D=256, 64 components.
// Strategy: bucket samples by component, then per-component GEMM tiles using
// the native f32 matrix op V_WMMA_F32_16X16X4_F32 (exact f32 semantics).

typedef __attribute__((ext_vector_type(2))) float v2f;
typedef __attribute__((ext_vector_type(8))) float v8f;

#define NG     64
#define DIM    256
#define BATCH  4096
#define MAXT   (BATCH / 16 + NG)   // 320 tiles max (256 full + up to 64 partial)

// ---------------------------------------------------------------------------
// Kernel 1: bucket samples by component k; pad buckets to multiples of 16
// with -1 sentinels. Produces:
//   tile_k[t]        : component of 16-sample tile t (-1 = unused tile)
//   padded_idx[t*16+i]: sample id in slot i of tile t (-1 = padding)
// Intra-bucket order from atomics is nondeterministic, but every sample's
// output row is computed identically and written to its own location, so the
// final d_out is deterministic.
// ---------------------------------------------------------------------------
__global__ __launch_bounds__(1024) void bucket_kernel(const int* __restrict__ k,
                                                      int* __restrict__ tile_k,
                                                      int* __restrict__ padded_idx) {
    __shared__ int s_cnt[NG];
    __shared__ int s_cur[NG];
    __shared__ int s_off[NG];
    const int tid = threadIdx.x;

    if (tid < NG) { s_cnt[tid] = 0; s_cur[tid] = 0; }
    for (int i = tid; i < MAXT; i += 1024)      tile_k[i] = -1;
    for (int i = tid; i < MAXT * 16; i += 1024) padded_idx[i] = -1;
    __syncthreads();

    for (int i = tid; i < BATCH; i += 1024)
        atomicAdd(&s_cnt[k[i]], 1);
    __syncthreads();

    if (tid == 0) {
        int off = 0;
        for (int c = 0; c < NG; ++c) {
            s_off[c] = off;
            const int nt = (s_cnt[c] + 15) >> 4;      // tiles for component c
            for (int t = 0; t < nt; ++t) tile_k[(off >> 4) + t] = c;
            off += nt << 4;
        }
    }
    __syncthreads();

    for (int i = tid; i < BATCH; i += 1024) {
        const int c   = k[i];
        const int pos = atomicAdd(&s_cur[c], 1);
        padded_idx[s_off[c] + pos] = i;
    }
}

// ---------------------------------------------------------------------------
// Kernel 2: one wave computes one 16(sample) x 16(dim) output tile.
// Block = 256 threads = 8 waves, gridDim.y = 2  ->  8*16*2 = 256 dims covered.
// K-loop: 64 x V_WMMA_F32_16X16X4_F32 over the D=256 contraction.
//
// Operand layout (32-bit WMMA, wave32): for both A(16x4) and B(4x16),
// lane L, VGPR v holds element [row_or_col = L%16, K = 2*(L/16)+v], i.e. each
// lane loads one contiguous float2 per step. C/D: lane L, VGPR v holds
// D[M = v + 8*(L/16), N = L%16].
// ---------------------------------------------------------------------------
__global__ __launch_bounds__(256) void gemm_kernel(const float* __restrict__ mu,
                                                   const float* __restrict__ A,
                                                   const float* __restrict__ eps,
                                                   const int*   __restrict__ tile_k,
                                                   const int*   __restrict__ padded_idx,
                                                   float*       __restrict__ out) {
    const int t  = blockIdx.x;
    const int kc = tile_k[t];
    if (kc < 0) return;                     // uniform per block: whole tile unused

    const int wave = threadIdx.x >> 5;
    const int lane = threadIdx.x & 31;
    const int half = lane >> 4;             // 0: K pair {0,1}; 1: K pair {2,3}
    const int col  = lane & 15;             // N (dim) column / M (sample) row id
    const int n0   = blockIdx.y * 128 + wave * 16;

    const int* tidx = padded_idx + t * 16;
    const int  sA   = tidx[col];            // sample feeding this lane's A rows
    // Sentinel rows read sample 0's eps (valid memory); results never stored.
    const float* epsRow = eps + (size_t)(sA >= 0 ? sA : 0) * DIM;
    // B operand: row (n0+col) of the component matrix A[kc].
    const float* Arow   = A + ((size_t)kc * DIM + (size_t)(n0 + col)) * DIM;

    const int koff = half * 2;
    v8f acc = {};

#pragma unroll 8
    for (int j0 = 0; j0 < DIM; j0 += 4) {
        v2f a = *(const v2f*)(epsRow + j0 + koff);   // eps[s, j0+koff .. +1]
        v2f b = *(const v2f*)(Arow  + j0 + koff);    // A[kc, n0+col, j0+koff .. +1]
        // D = A x B + C, exact f32: (neg_a, A, neg_b, B, c_mod, C, reuse_a, reuse_b)
        acc = __builtin_amdgcn_wmma_f32_16x16x4_f32(
            false, a, false, b, (short)0, acc, false, false);
    }

    const float m = mu[kc * DIM + n0 + col];
#pragma unroll
    for (int v = 0; v < 8; ++v) {
        const int sD = tidx[v + 8 * half];           // sample for D row M=v+8*half
        if (sD >= 0)
            out[(size_t)sD * DIM + n0 + col] = acc[v] + m;
    }
}

// ---------------------------------------------------------------------------
extern "C" void kernel_launch(void* const* d_in, const int* in_sizes, int n_in,
                              void* d_out, int out_size, void* d_ws, size_t ws_size,
                              hipStream_t stream) {
    const float* mu  = (const float*)d_in[0];   // [64, 256]
    const float* A   = (const float*)d_in[1];   // [64, 256, 256]
    const float* eps = (const float*)d_in[2];   // [4096, 256]
    const int*   k   = (const int*)d_in[3];     // [4096]
    float* out = (float*)d_out;                 // [4096, 256]

    int* tile_k     = (int*)d_ws;               // MAXT ints
    int* padded_idx = tile_k + MAXT;            // MAXT*16 ints

    bucket_kernel<<<1, 1024, 0, stream>>>(k, tile_k, padded_idx);
    gemm_kernel<<<dim3(MAXT, 2, 1), dim3(256, 1, 1), 0, stream>>>(
        mu, A, eps, tile_k, padded_idx, out);
}